// PatchShuffle_26869315404121
// MI455X (gfx1250) — compile-verified
//
#include <hip/hip_runtime.h>
#include <cstdint>
#include <cstddef>

// PatchShuffle for x[16,64,256,256] f32, PH=PW=2, shuffle_prob=0.1
// HBM-bound streaming kernel (roofline: 512 MiB @ 23.3 TB/s ~ 22 us).
// CDNA5 paths: async global->LDS (ASYNCcnt) quad-buffered pipeline with
// prefetch depth 2, B128 everywhere, NT load/store temporal hints,
// wave32-friendly 256-thread (8-wave) blocks, one barrier per 16 KB tile.

typedef float v4f __attribute__((ext_vector_type(4)));

constexpr int Bn = 16, Cn = 64, Hn = 256, Wn = 256;
constexpr int TILE_ROWS = 16;                // 8 row-pairs per tile
constexpr int W4        = Wn / 4;            // 64 float4 per row
constexpr int TILE_F4   = TILE_ROWS * W4;    // 1024 float4 = 16 KB per tile
constexpr int TILES     = Hn / TILE_ROWS;    // 16 tiles per (b,c) plane
constexpr int THREADS   = 256;               // 8 wave32
constexpr int NBUF      = 4;                 // 64 KB LDS, issue depth 2

__device__ __forceinline__ unsigned hash32(unsigned x) {
  x ^= x >> 16; x *= 0x7feb352du;
  x ^= x >> 15; x *= 0x846ca68bu;
  x ^= x >> 16;
  return x;
}

// Uniform random permutation of 4 pixels: sort (random-key, pixel) pairs with a
// 5-comparator network == take_along_axis(argsort(keys)). Keys index-tagged so
// ties resolve stably, matching argsort semantics.
__device__ __forceinline__ void shuffle_patch(unsigned h,
                                              float& a, float& b,
                                              float& c, float& d) {
  unsigned k0 = ((h        & 0xFFu) << 2) | 0u;
  unsigned k1 = (((h >> 8) & 0xFFu) << 2) | 1u;
  unsigned k2 = (((h >>16) & 0xFFu) << 2) | 2u;
  unsigned k3 = (((h >>24) & 0xFFu) << 2) | 3u;
  float v0 = a, v1 = b, v2 = c, v3 = d;
#define CSWAP_(ki,kj,vi,vj)                                                  \
  { bool s_ = (kj) < (ki); unsigned tk_ = ki; float tv_ = vi;                \
    ki = s_ ? kj : ki; kj = s_ ? tk_ : kj;                                   \
    vi = s_ ? vj : vi; vj = s_ ? tv_ : vj; }
  CSWAP_(k0, k1, v0, v1)
  CSWAP_(k2, k3, v2, v3)
  CSWAP_(k0, k2, v0, v2)
  CSWAP_(k1, k3, v1, v3)
  CSWAP_(k1, k2, v1, v2)
#undef CSWAP_
  a = v0; b = v1; c = v2; d = v3;
}

__global__ __launch_bounds__(THREADS, 2)
void patchshuffle_kernel(const float* __restrict__ xg, float* __restrict__ og) {
  __shared__ v4f tile[NBUF][TILE_F4];        // 4 x 16 KB ring buffer

  const int plane = blockIdx.x;              // b*C + c  (1024 planes)
  const int bidx  = plane >> 6;              // sample index b
  // per-sample Bernoulli(0.1): block-uniform -> no divergence
  const bool doShuffle =
      hash32(0x9E3779B9u * (unsigned)bidx + 0x85EBCA6Bu) < 429496729u;

  const int t = threadIdx.x;
  const v4f* __restrict__ src = (const v4f*)(xg + (size_t)plane * (Hn * Wn));
  v4f* __restrict__ dst       = (v4f*)(og + (size_t)plane * (Hn * Wn));

  const int rpb = t >> 6;                    // base row-pair (0..3)
  const int xq  = t & 63;                    // float4 column (2 patches)

  // 4 async B128 issues per wave per tile; input is touch-once -> NT
  auto issue = [&](int tl, int buf) {
    const v4f* gbase = src + tl * TILE_F4;
#pragma unroll
    for (int k = 0; k < 4; ++k) {
      const int e = t + THREADS * k;         // 1024 elements per tile
      unsigned l = (unsigned)(uintptr_t)&tile[buf][e];
      asm volatile("global_load_async_to_lds_b128 %0, %1, off th:TH_LOAD_NT"
                   :: "v"(l), "v"(gbase + e) : "memory");
    }
  };

  issue(0, 0);
  issue(1, 1);

  for (int tl = 0; tl < TILES; ++tl) {
    const int buf = tl & (NBUF - 1);
    if (tl + 2 < TILES) {
      // overwrites buf (tl+2)%4, last read at iter tl-2: protected by the
      // barrier of iter tl-1 (D=2 <= NBUF-2).
      issue(tl + 2, (tl + 2) & (NBUF - 1));
      // 12 async ops outstanding; <=8 pending means tile tl has landed
      asm volatile("s_wait_asynccnt 0x8" ::: "memory");
    } else if (tl + 1 < TILES) {
      asm volatile("s_wait_asynccnt 0x4" ::: "memory");
    } else {
      asm volatile("s_wait_asynccnt 0x0" ::: "memory");
    }
    __syncthreads();   // single barrier per tile: data-ready + WAR protection

#pragma unroll
    for (int u = 0; u < 2; ++u) {
      const int rp = rpb + 4 * u;            // row-pair within tile (0..7)
      v4f top = tile[buf][(2 * rp    ) * W4 + xq];   // ds_load_b128
      v4f bot = tile[buf][(2 * rp + 1) * W4 + xq];

      if (doShuffle) {
        const unsigned hp = (unsigned)(tl * 8 + rp);             // 0..127
        const unsigned pb =
            ((unsigned)plane * 128u + hp) * 128u + 2u * (unsigned)xq;
        float a, b, c, d;
        a = top.x; b = top.y; c = bot.x; d = bot.y;              // patch A
        shuffle_patch(hash32(pb), a, b, c, d);
        top.x = a; top.y = b; bot.x = c; bot.y = d;
        a = top.z; b = top.w; c = bot.z; d = bot.w;              // patch B
        shuffle_patch(hash32(pb + 1u), a, b, c, d);
        top.z = a; top.w = b; bot.z = c; bot.w = d;
      }

      const int row0 = tl * TILE_ROWS + 2 * rp;
      // streaming output: non-temporal B128 stores
      __builtin_nontemporal_store(top, &dst[(row0    ) * W4 + xq]);
      __builtin_nontemporal_store(bot, &dst[(row0 + 1) * W4 + xq]);
    }
  }
}

extern "C" void kernel_launch(void* const* d_in, const int* in_sizes, int n_in,
                              void* d_out, int out_size, void* d_ws,
                              size_t ws_size, hipStream_t stream) {
  (void)in_sizes; (void)n_in; (void)out_size; (void)d_ws; (void)ws_size;
  const float* x = (const float*)d_in[0];
  float* out = (float*)d_out;
  patchshuffle_kernel<<<dim3(Bn * Cn), dim3(THREADS), 0, stream>>>(x, out);
}